// LabelSmoothingCrossEntropy_53609781789400
// MI455X (gfx1250) — compile-verified
//
#include <hip/hip_runtime.h>
#include <math.h>

#define NCLS 8192
#define NROW 16384
#define SMOOTH 0.1f

// ---------- CDNA5 async global->LDS copy (ASYNCcnt-tracked, fire-and-forget) ----------
__device__ __forceinline__ void async_g2l_b128(unsigned int lds_byte_off, const float* gaddr) {
#if defined(__gfx1250__)
    // VDST = VGPR holding LDS byte address, VADDR = VGPR pair with 64-bit global address (GV mode)
    asm volatile("global_load_async_to_lds_b128 %0, %1, off"
                 :: "v"(lds_byte_off), "v"((unsigned long long)(size_t)gaddr)
                 : "memory");
#else
    (void)lds_byte_off; (void)gaddr;
#endif
}

// Wait until ASYNCcnt <= n. Async loads complete in order, so after this the
// first (8-n) issued chunks are guaranteed resident in LDS.
__device__ __forceinline__ void wait_async_le(int n) {
#if defined(__gfx1250__)
    switch (n) {   // constant-folds under full unroll
        case 0: asm volatile("s_wait_asynccnt 0x0" ::: "memory"); break;
        case 1: asm volatile("s_wait_asynccnt 0x1" ::: "memory"); break;
        case 2: asm volatile("s_wait_asynccnt 0x2" ::: "memory"); break;
        case 3: asm volatile("s_wait_asynccnt 0x3" ::: "memory"); break;
        case 4: asm volatile("s_wait_asynccnt 0x4" ::: "memory"); break;
        case 5: asm volatile("s_wait_asynccnt 0x5" ::: "memory"); break;
        case 6: asm volatile("s_wait_asynccnt 0x6" ::: "memory"); break;
        default: asm volatile("s_wait_asynccnt 0x7" ::: "memory"); break;
    }
#else
    (void)n;
#endif
}

// ---------- Kernel 1: per-row statistics (one workgroup per row) ----------
__global__ __launch_bounds__(256) void lsce_rowstats_kernel(
    const float* __restrict__ pred,
    const int*   __restrict__ target,
    float4*      __restrict__ stats)   // per row: {lse, 1/Z, rowsum, pred[target]}
{
    __shared__ float sbuf[NCLS];           // 32 KB row staging buffer
    __shared__ float rmax[8], rsum[8], rexp[8];

    const int tid  = threadIdx.x;
    const int row  = blockIdx.x;
    const int lane = tid & 31;
    const int wid  = tid >> 5;
    const float* base = pred + (size_t)row * NCLS;

    // Gather pred[row, target[row]] early (thread 0 only) to hide its latency.
    float predt = 0.0f;
    if (tid == 0) predt = base[target[row]];

    // Stage the full row into LDS with fire-and-forget async b128 copies.
    // Round r covers columns [r*1024, r*1024+1024); thread t handles 4 floats.
#pragma unroll
    for (int r = 0; r < 8; ++r) {
        const int col = r * 1024 + tid * 4;
        // generic pointer to __shared__ has the LDS byte offset in its low 32 bits
        const unsigned int loff = (unsigned int)(size_t)(&sbuf[col]);
        async_g2l_b128(loff, base + col);
    }

    // Pass 1: consume chunks as they arrive (each lane reads back only the
    // bytes its own async op wrote, so per-wave ASYNCcnt waits suffice and
    // no workgroup barrier is needed for the staged data).
    float v[32];
    float tmax = -INFINITY;
    float tsum = 0.0f;
#pragma unroll
    for (int r = 0; r < 8; ++r) {
        wait_async_le(7 - r);
        const float4 q = *(const float4*)&sbuf[r * 1024 + tid * 4];
        v[r * 4 + 0] = q.x; v[r * 4 + 1] = q.y; v[r * 4 + 2] = q.z; v[r * 4 + 3] = q.w;
        tmax = fmaxf(tmax, fmaxf(fmaxf(q.x, q.y), fmaxf(q.z, q.w)));
        tsum += (q.x + q.y) + (q.z + q.w);
    }
    // wave32 butterfly reduction
#pragma unroll
    for (int o = 16; o > 0; o >>= 1) {
        tmax = fmaxf(tmax, __shfl_xor(tmax, o, 32));
        tsum += __shfl_xor(tsum, o, 32);
    }
    if (lane == 0) { rmax[wid] = tmax; rsum[wid] = tsum; }
    __syncthreads();

    // All threads redundantly combine the 8 wave partials (serves as broadcast).
    float m  = rmax[0];
    float rs = rsum[0];
#pragma unroll
    for (int i = 1; i < 8; ++i) { m = fmaxf(m, rmax[i]); rs += rsum[i]; }

    // Pass 2: sum of exp(v - m), values already resident in VGPRs.
    float esum = 0.0f;
#pragma unroll
    for (int i = 0; i < 32; ++i) esum += __expf(v[i] - m);
#pragma unroll
    for (int o = 16; o > 0; o >>= 1) esum += __shfl_xor(esum, o, 32);
    if (lane == 0) rexp[wid] = esum;
    __syncthreads();

    if (tid == 0) {
        float Z = rexp[0];
#pragma unroll
        for (int i = 1; i < 8; ++i) Z += rexp[i];
        const float lse = m + __logf(Z);
        // max softmax prob of the row == exp(m - lse) == 1/Z
        stats[row] = make_float4(lse, 1.0f / Z, rs, predt);
    }
}

// ---------- Kernel 2: global reduction + final scalar ----------
__global__ __launch_bounds__(1024) void lsce_finalize_kernel(
    const float4* __restrict__ stats,
    const int*    __restrict__ target,
    const float*  __restrict__ cw,
    float*        __restrict__ out)
{
    __shared__ float red[32];
    __shared__ float s_smooth;

    const int tid  = threadIdx.x;
    const int lane = tid & 31;
    const int wid  = tid >> 5;

    // Phase A: mean of 1/Z  ->  adaptive smoothing
    float acc = 0.0f;
    for (int i = tid; i < NROW; i += 1024) acc += stats[i].y;
#pragma unroll
    for (int o = 16; o > 0; o >>= 1) acc += __shfl_xor(acc, o, 32);
    if (lane == 0) red[wid] = acc;
    __syncthreads();
    if (tid == 0) {
        float s = 0.0f;
#pragma unroll
        for (int i = 0; i < 32; ++i) s += red[i];
        s_smooth = SMOOTH * (1.0f - s / (float)NROW);
    }
    __syncthreads();

    const float smoothing = s_smooth;
    const float conf  = 1.0f - smoothing;
    const float offv  = smoothing / (float)(NCLS - 1);
    const float cmo   = conf - offv;

    // Phase B: weighted per-row losses
    float lacc = 0.0f;
    for (int i = tid; i < NROW; i += 1024) {
        const float4 st = stats[i];                  // {lse, 1/Z, rowsum, predt}
        const float S = st.z - (float)NCLS * st.x;   // sum_j logp = rowsum - C*lse
        const float t = st.w - st.x;                 // logp at target
        lacc += -(offv * S + cmo * t) * cw[target[i]];
    }
#pragma unroll
    for (int o = 16; o > 0; o >>= 1) lacc += __shfl_xor(lacc, o, 32);
    if (lane == 0) red[wid] = lacc;
    __syncthreads();
    if (tid == 0) {
        float s = 0.0f;
#pragma unroll
        for (int i = 0; i < 32; ++i) s += red[i];
        out[0] = s / (float)NROW;
    }
}

extern "C" void kernel_launch(void* const* d_in, const int* in_sizes, int n_in,
                              void* d_out, int out_size, void* d_ws, size_t ws_size,
                              hipStream_t stream) {
    const float* pred   = (const float*)d_in[0];
    const int*   target = (const int*)d_in[1];
    const float* cw     = (const float*)d_in[2];
    float4* stats = (float4*)d_ws;          // 16384 * 16 B = 256 KB scratch

    lsce_rowstats_kernel<<<NROW, 256, 0, stream>>>(pred, target, stats);
    lsce_finalize_kernel<<<1, 1024, 0, stream>>>(stats, target, cw, (float*)d_out);
}